// InductiveRadialBiasingForce_10823317586375
// MI455X (gfx1250) — compile-verified
//
#include <hip/hip_runtime.h>
#include <math.h>

#define NPART 1024
#define NCH   64          // NUM_OFFSETS * NUM_ORDERS
#define TILE  16
#define NTILE (NPART / TILE)   // 64

typedef __attribute__((ext_vector_type(2))) float v2f;
typedef __attribute__((ext_vector_type(4))) float v4f;
typedef __attribute__((ext_vector_type(8))) float v8f;

// ---------------------------------------------------------------------------
// Prep: per (particle i, channel ch) build WMMA fragments for the rank-3
// bilinear exponent  -(u_j - b_i)^2 * p_i * q_j = A_i . B_j  (K=4, last=0)
//   A_i = log2e * (-p, 2 b p, -b^2 p, 0)     b = offset_i, p = gamma_i^2
//   B_j =         (u^2 q, u q, q, 0)         u = r_j - offset_j, q = gamma_j^2
// and fold t_r into the coefficient: c' = coeff * sin(2*pi*r*t).
// Storage [ch][i][4] so each lane's b64 load is its A/B fragment half:
// lanes 0-15 -> (K0,K1), lanes 16-31 -> (K2,K3)  (ISA 32-bit A layout).
// ---------------------------------------------------------------------------
__global__ void irbf_prep(const float* __restrict__ x,
                          const float* __restrict__ t,
                          const float* __restrict__ offset,
                          const float* __restrict__ gamma,
                          const float* __restrict__ coeff,
                          float* __restrict__ Afrag,   // [NCH][NPART][4]
                          float* __restrict__ Bfrag,   // [NCH][NPART][4]
                          float* __restrict__ Cp,      // [NCH][NPART]
                          float* __restrict__ out) {
  int tid = blockIdx.x * blockDim.x + threadIdx.x;     // i*NCH + ch
  if (tid == 0) out[0] = 0.0f;                         // zero accumulator
  if (tid >= NPART * NCH) return;

  int i  = tid / NCH;
  int ch = tid - i * NCH;        // ch = o*8 + r  (matches input flat layout)
  int r  = ch & 7;

  float tv = __sinf(t[0] * 6.28318530717958647692f * (float)r);

  float x0 = x[i * 3 + 0], x1 = x[i * 3 + 1], x2 = x[i * 3 + 2];
  float ri = sqrtf(x0 * x0 + x1 * x1 + x2 * x2);

  float b = offset[tid];
  float g = gamma[tid];
  float p = g * g;
  const float LOG2E = 1.44269504088896340736f;
  float pl = p * LOG2E;

  int idx = (ch * NPART + i) * 4;
  Afrag[idx + 0] = -pl;
  Afrag[idx + 1] = 2.0f * b * pl;
  Afrag[idx + 2] = -b * b * pl;
  Afrag[idx + 3] = 0.0f;

  float u = ri - b;
  Bfrag[idx + 0] = u * u * p;
  Bfrag[idx + 1] = u * p;
  Bfrag[idx + 2] = p;
  Bfrag[idx + 3] = 0.0f;

  Cp[ch * NPART + i] = coeff[tid] * tv;
}

// ---------------------------------------------------------------------------
// Main: one wave per 16x16 (i,j) tile; loop over 64 channels.
// Per channel: 1x v_wmma_f32_16x16x4_f32 -> 256 exponents (log2 units) in
// 8 VGPRs, then v_exp_f32 + pk_add weight (c'_i + c'_j) + pk_fma into an
// 8-wide vector accumulator (independent chains; no serial scalar adds).
// C/D layout: VGPR v, lanes 0-15: (M=v,   N=lane),
//                     lanes 16-31:(M=v+8, N=lane-16).
// ---------------------------------------------------------------------------
__global__ void irbf_main(const float* __restrict__ Afrag,
                          const float* __restrict__ Bfrag,
                          const float* __restrict__ Cp,
                          float* __restrict__ out) {
  int lane = threadIdx.x & 31;
  int w    = blockIdx.x * (blockDim.x >> 5) + (threadIdx.x >> 5); // 0..4095
  int ti   = w / NTILE;           // i-tile
  int tj   = w - ti * NTILE;      // j-tile
  int sel  = lane >> 4;           // 0: K0/K1  1: K2/K3
  int m    = lane & 15;           // row (A) / col (B) within tile

  float vacc[8];
#pragma unroll
  for (int v = 0; v < 8; ++v) vacc[v] = 0.0f;

  for (int ch = 0; ch < NCH; ++ch) {
    const float* ap = Afrag + ((size_t)(ch * NPART + ti * TILE + m) * 4 + sel * 2);
    const float* bp = Bfrag + ((size_t)(ch * NPART + tj * TILE + m) * 4 + sel * 2);
    v2f a = *(const v2f*)ap;
    v2f b = *(const v2f*)bp;

    v8f d = {};
    // D = A x B : D[m][n] = exponent(i = ti*16+m, j = tj*16+n), in log2 units
    d = __builtin_amdgcn_wmma_f32_16x16x4_f32(
        /*neg_a=*/false, a, /*neg_b=*/false, b,
        /*c_mod=*/(short)0, d, /*reuse_a=*/false, /*reuse_b=*/false);

    // c'_j for this lane's column (per-lane), c'_i for the 8 row indices
    float cj = Cp[ch * NPART + tj * TILE + m];
    const float* cib = Cp + ch * NPART + ti * TILE + sel * 8;
    v4f ci0 = *(const v4f*)(cib);
    v4f ci1 = *(const v4f*)(cib + 4);

    float wgt[8];
#pragma unroll
    for (int v = 0; v < 4; ++v) { wgt[v] = ci0[v] + cj; wgt[v + 4] = ci1[v] + cj; }

#pragma unroll
    for (int v = 0; v < 8; ++v)
      vacc[v] = fmaf(wgt[v], __builtin_amdgcn_exp2f(d[v]), vacc[v]);
  }

  // fold the 8-wide accumulator, then wave32 shuffle reduction
  float local = ((vacc[0] + vacc[1]) + (vacc[2] + vacc[3])) +
                ((vacc[4] + vacc[5]) + (vacc[6] + vacc[7]));
#pragma unroll
  for (int off = 16; off >= 1; off >>= 1)
    local += __shfl_xor(local, off, 32);

  if (lane == 0) atomicAdd(out, local);
}

// ---------------------------------------------------------------------------
extern "C" void kernel_launch(void* const* d_in, const int* in_sizes, int n_in,
                              void* d_out, int out_size, void* d_ws, size_t ws_size,
                              hipStream_t stream) {
  const float* x      = (const float*)d_in[0];
  const float* t      = (const float*)d_in[1];
  const float* offset = (const float*)d_in[2];
  const float* gamma  = (const float*)d_in[3];
  const float* coeff  = (const float*)d_in[4];
  float* out = (float*)d_out;

  float* Afrag = (float*)d_ws;                     // NCH*NPART*4 floats (1 MB)
  float* Bfrag = Afrag + (size_t)NCH * NPART * 4;  // 1 MB
  float* Cp    = Bfrag + (size_t)NCH * NPART * 4;  // 256 KB

  // prep: 65536 threads (also zeroes d_out; same-stream order guarantees it
  // completes before the main kernel's atomics)
  irbf_prep<<<(NPART * NCH) / 256, 256, 0, stream>>>(
      x, t, offset, gamma, coeff, Afrag, Bfrag, Cp, out);

  // main: 64x64 tiles = 4096 waves, 8 waves per 256-thread block
  irbf_main<<<(NTILE * NTILE) / 8, 256, 0, stream>>>(Afrag, Bfrag, Cp, out);
}